// GCN_37821482008889
// MI455X (gfx1250) — compile-verified
//
#include <hip/hip_runtime.h>

typedef __attribute__((ext_vector_type(2))) float v2f;
typedef __attribute__((ext_vector_type(8))) float v8f;

#define DFEAT 32
#define DCOMB 160
#define DHID  128
#define MT    10          // M-tiles (16 rows each) per block in the GEMMs

// ---------------- utility ----------------
__global__ void zero_kernel(float* __restrict__ p, long n) {
  long t = (long)blockIdx.x * blockDim.x + threadIdx.x;
  if (t < n) p[t] = 0.0f;
}

// ---------------- build z = [emb | emb[neigh]] ----------------
__global__ void build_z_kernel(const int* __restrict__ nodes,
                               const int* __restrict__ neigh,
                               const float* __restrict__ emb,
                               float* __restrict__ z, int N) {
  int t = blockIdx.x * blockDim.x + threadIdx.x;
  int total = N * (DCOMB / 4);
  if (t >= total) return;
  int i  = t / (DCOMB / 4);
  int c4 = t - i * (DCOMB / 4);     // 0..39
  int srcRow;
  if (c4 < 8) srcRow = nodes[i];                       // own features
  else        srcRow = neigh[i * 4 + ((c4 - 8) >> 3)]; // neighbour j
  int c = c4 & 7;
  float4 v = ((const float4*)(emb + (long)srcRow * DFEAT))[c];
  ((float4*)(z + (long)i * DCOMB))[c4] = v;
}

// ---------------- degree / dinv / norm ----------------
__global__ void deg_accum_kernel(const int* __restrict__ dst,
                                 const float* __restrict__ w,
                                 float* __restrict__ deg, int E) {
  int t = blockIdx.x * blockDim.x + threadIdx.x;
  if (t < E) unsafeAtomicAdd(&deg[dst[t]], w[t]);
}

__global__ void dinv_kernel(float* __restrict__ dinv, int N) {
  int t = blockIdx.x * blockDim.x + threadIdx.x;
  if (t < N) dinv[t] = rsqrtf(dinv[t] + 1.0f);   // self-loop weight 1 => deg > 0
}

__global__ void norm_kernel(const int* __restrict__ src, const int* __restrict__ dst,
                            const float* __restrict__ w, const float* __restrict__ dinv,
                            float* __restrict__ norm, int E) {
  int t = blockIdx.x * blockDim.x + threadIdx.x;
  if (t < E) norm[t] = dinv[src[t]] * w[t] * dinv[dst[t]];
}

// ---------------- LDS-staged fp32 WMMA GEMM: C[M,N] = A[M,K] @ B[K,N] ----------
// grid.x blocks each own MT consecutive 16-row M-tiles. blockDim = 32*(N/16),
// wave w owns N-tile w. All of B (the weight matrix, <=80 KB) is staged in LDS
// once per block; each 16xK A-tile is staged in LDS and shared by all waves.
// Fragment layout per ISA 7.12.2 (16x4 f32): lanes 0-15 hold K=0,1; lanes
// 16-31 hold K=2,3. C/D: VGPR r -> row 8*(lane>>4)+r, col = lane&15.
__global__ void gemm_wmma_lds(const float* __restrict__ A, const float* __restrict__ B,
                              float* __restrict__ C, int K, int N) {
  extern __shared__ float smem[];
  float* sW = smem;            // K*N floats
  float* sA = smem + K * N;    // 16*K floats

  const int tid  = threadIdx.x;
  const int bdx  = blockDim.x;
  const int lane = tid & 31;
  const int wave = tid >> 5;
  const int l15  = lane & 15;
  const int hi   = lane >> 4;      // 0 or 1
  const int koff = hi << 1;        // 0 or 2
  const int n0   = wave << 4;

  // stage whole weight matrix into LDS (float4 wide)
  const int wq = (K * N) >> 2;
  for (int i = tid; i < wq; i += bdx)
    ((float4*)sW)[i] = ((const float4*)B)[i];
  __syncthreads();

  const int kq = K >> 2;           // float4s per A row
  for (int t = 0; t < MT; ++t) {
    const int m0 = (blockIdx.x * MT + t) << 4;

    // stage 16xK A-tile into LDS
    for (int i = tid; i < 16 * kq; i += bdx) {
      int r = i / kq, c = i - r * kq;
      ((float4*)sA)[i] = ((const float4*)(A + (long)(m0 + r) * K))[c];
    }
    // prefetch next A-tile (global_prefetch_b8) while we compute this one
    if (t + 1 < MT) {
      const char* nA = (const char*)(A + (long)(m0 + 16) * K);
      for (int i = tid; i < K; i += bdx)          // 16*K*4/64 = K cachelines
        __builtin_prefetch(nA + (long)i * 64, 0, 0);
    }
    __syncthreads();

    const float* saRow = sA + l15 * K;
    v8f acc = {};
    for (int k = 0; k < K; k += 4) {
      v2f a, b;
      a.x = saRow[k + koff];                       // ds_load_b64 pair
      a.y = saRow[k + koff + 1];
      b.x = sW[(k + koff) * N + n0 + l15];
      b.y = sW[(k + koff + 1) * N + n0 + l15];
      acc = __builtin_amdgcn_wmma_f32_16x16x4_f32(false, a, false, b,
                                                  (short)0, acc, false, false);
    }

    float* crow = C + (long)(m0 + (hi << 3)) * N + n0 + l15;
#pragma unroll
    for (int r = 0; r < 8; ++r)
      crow[(long)r * N] = acc[r];
    __syncthreads();   // sA reused next iteration
  }
}

// ---------------- edge message scatter: agg[dst] += h[src] * norm ----------------
__global__ void scatter_kernel(const int* __restrict__ src, const int* __restrict__ dst,
                               const float* __restrict__ norm, const float* __restrict__ h,
                               float* __restrict__ agg, int E, int D4 /* = D/4 */) {
  int t = blockIdx.x * blockDim.x + threadIdx.x;
  int total = E * D4;
  if (t >= total) return;
  int e  = t / D4;
  int c4 = t - e * D4;
  float nrm = norm[e];
  int s = src[e], d = dst[e];
  int D = D4 << 2;
  float4 v = ((const float4*)(h + (long)s * D))[c4];
  float* o = agg + (long)d * D + (c4 << 2);
  unsafeAtomicAdd(o + 0, v.x * nrm);
  unsafeAtomicAdd(o + 1, v.y * nrm);
  unsafeAtomicAdd(o + 2, v.z * nrm);
  unsafeAtomicAdd(o + 3, v.w * nrm);
}

// ---------------- finalize: agg = (relu?)(agg + h*dinv^2 + b) in place ----------------
__global__ void finalize_kernel(float* __restrict__ agg, const float* __restrict__ h,
                                const float* __restrict__ dinv, const float* __restrict__ b,
                                int N, int D, int doRelu) {
  int t = blockIdx.x * blockDim.x + threadIdx.x;
  long total = (long)N * D;
  if (t >= total) return;
  int i = t / D;
  int c = t - i * D;
  float di = dinv[i];
  float v = agg[t] + h[t] * (di * di) + b[c];
  agg[t] = doRelu ? fmaxf(v, 0.0f) : v;
}

// ---------------- edge decode: out[e] = relu(dot(z2[src], z2[dst])) ----------------
__global__ void edge_decode_kernel(const int* __restrict__ src, const int* __restrict__ dst,
                                   const float* __restrict__ z2, float* __restrict__ out,
                                   int E) {
  int wid  = (blockIdx.x * blockDim.x + threadIdx.x) >> 5;
  int lane = threadIdx.x & 31;
  if (wid >= E) return;
  const float* zs = z2 + (long)src[wid] * DCOMB;
  const float* zd = z2 + (long)dst[wid] * DCOMB;
  float acc = 0.0f;
#pragma unroll
  for (int j = 0; j < 5; ++j) {
    int c = lane + (j << 5);
    acc += zs[c] * zd[c];
  }
#pragma unroll
  for (int off = 16; off > 0; off >>= 1)
    acc += __shfl_xor(acc, off, 32);
  if (lane == 0) out[wid] = fmaxf(acc, 0.0f);
}

// ---------------- host ----------------
extern "C" void kernel_launch(void* const* d_in, const int* in_sizes, int n_in,
                              void* d_out, int out_size, void* d_ws, size_t ws_size,
                              hipStream_t stream) {
  const int*   nodes = (const int*)d_in[0];
  const int*   eidx  = (const int*)d_in[1];
  const float* ew    = (const float*)d_in[2];
  const int*   neigh = (const int*)d_in[3];
  const float* emb   = (const float*)d_in[4];
  const float* W1    = (const float*)d_in[5];
  const float* b1    = (const float*)d_in[6];
  const float* W2    = (const float*)d_in[7];
  const float* b2    = (const float*)d_in[8];
  float* out = (float*)d_out;

  const int N = in_sizes[0];
  const int E = in_sizes[2];
  const int* src = eidx;
  const int* dst = eidx + E;

  float* ws   = (float*)d_ws;
  float* dinv = ws;                         // N   (deg accumulator, then dinv)
  float* nrm  = dinv + N;                   // E
  float* z    = nrm + E;                    // N*160 (reused as h2 later)
  float* h1   = z    + (long)N * DCOMB;     // N*128
  float* agg1 = h1   + (long)N * DHID;      // N*128 -> z1 (in place)
  float* agg2 = agg1 + (long)N * DHID;      // N*160 -> z2 (in place)
  float* h2   = z;

  const int T = 256;
  auto blk = [](long n, int t) { return (int)((n + t - 1) / t); };

  // zero accumulators (fresh every call; graph-replay safe)
  zero_kernel<<<blk((long)N, T), T, 0, stream>>>(dinv, N);
  zero_kernel<<<blk((long)N * DHID, T), T, 0, stream>>>(agg1, (long)N * DHID);
  zero_kernel<<<blk((long)N * DCOMB, T), T, 0, stream>>>(agg2, (long)N * DCOMB);

  // gather features
  build_z_kernel<<<blk((long)N * (DCOMB / 4), T), T, 0, stream>>>(nodes, neigh, emb, z, N);

  // degree -> dinv -> per-edge norm
  deg_accum_kernel<<<blk(E, T), T, 0, stream>>>(dst, ew, dinv, E);
  dinv_kernel<<<blk(N, T), T, 0, stream>>>(dinv, N);
  norm_kernel<<<blk(E, T), T, 0, stream>>>(src, dst, ew, dinv, nrm, E);

  // layer 1: h1 = z @ W1  (K=160 -> 128) — LDS-staged WMMA f32
  {
    int grid = N / (16 * MT);                              // 625
    size_t shm = (size_t)(DCOMB * DHID + 16 * DCOMB) * sizeof(float);
    gemm_wmma_lds<<<grid, 32 * (DHID / 16), shm, stream>>>(z, W1, h1, DCOMB, DHID);
  }
  scatter_kernel<<<blk((long)E * (DHID / 4), T), T, 0, stream>>>(src, dst, nrm, h1, agg1, E, DHID / 4);
  finalize_kernel<<<blk((long)N * DHID, T), T, 0, stream>>>(agg1, h1, dinv, b1, N, DHID, 1);

  // layer 2: h2 = z1 @ W2  (K=128 -> 160) — LDS-staged WMMA f32
  {
    int grid = N / (16 * MT);
    size_t shm = (size_t)(DHID * DCOMB + 16 * DHID) * sizeof(float);
    gemm_wmma_lds<<<grid, 32 * (DCOMB / 16), shm, stream>>>(agg1, W2, h2, DHID, DCOMB);
  }
  scatter_kernel<<<blk((long)E * (DCOMB / 4), T), T, 0, stream>>>(src, dst, nrm, h2, agg2, E, DCOMB / 4);
  finalize_kernel<<<blk((long)N * DCOMB, T), T, 0, stream>>>(agg2, h2, dinv, b2, N, DCOMB, 0);

  // edge decode: one wave per edge
  edge_decode_kernel<<<blk((long)E * 32, T), T, 0, stream>>>(src, dst, agg2, out, E);
}